// Linear_thetafunc_56719338111702
// MI455X (gfx1250) — compile-verified
//
#include <hip/hip_runtime.h>

#define BATCH   256
#define WINDOW  32
#define SIZE    512
#define PLENGTH 128

typedef __attribute__((ext_vector_type(2))) float v2f;
typedef __attribute__((ext_vector_type(8))) float v8f;
typedef __attribute__((ext_vector_type(4))) unsigned int v4u;
typedef __attribute__((ext_vector_type(4))) int v4i;
typedef __attribute__((ext_vector_type(8))) int v8i;

#if defined(__has_builtin)
#if __has_builtin(__builtin_amdgcn_tensor_load_to_lds)
#define USE_TDM 1
#pragma message("path: TDM tensor_load_to_lds")
#endif
#endif
#ifndef USE_TDM
#define USE_TDM 0
#pragma message("path: inline-asm global_load_async_to_lds_b128 fallback")
#endif

// ---------------------------------------------------------------------------
// Fallback: async global -> LDS copy, 16 bytes per lane (GV form).
// ---------------------------------------------------------------------------
__device__ __forceinline__ void async_cp16(const void* g, void* lds) {
    asm volatile("global_load_async_to_lds_b128 %0, %1, off"
                 :: "v"((uint32_t)(uintptr_t)lds), "v"((uint64_t)(uintptr_t)g)
                 : "memory");
}

#if USE_TDM
// ---------------------------------------------------------------------------
// TDM: DMA a 2D tile (tile_d0 cols x tile_d1 rows, f32, row stride in floats)
// from global to LDS, padding each 64-DWORD row by pad_amount_dw DWORDs so the
// LDS pitch becomes 64+pad (bank-conflict-free fragment reads).
// D# packing per CDNA5 ISA 8.3/8.4; groups 2/3 zero (2-D tensor).
// ---------------------------------------------------------------------------
__device__ __forceinline__ void tdm_load_2d(const void* gptr, void* ldsptr,
                                            unsigned tile_d0, unsigned tile_d1,
                                            unsigned stride_elts,
                                            unsigned pad_amount_dw) {
    unsigned long long ga = (unsigned long long)(uintptr_t)gptr;
    v4u g0;
    g0.x = 1u;                                            // count=1, user mode
    g0.y = (unsigned)(uintptr_t)ldsptr;                   // lds_addr (bytes)
    g0.z = (unsigned)(ga & 0xFFFFFFFFu);                  // global_addr[31:0]
    g0.w = (unsigned)((ga >> 32) & 0x01FFFFFFu)           // global_addr[56:32]
         | 0x80000000u;                                   // type=2 ("image")

    v8i g1;
    g1[0] = (int)((2u << 16)                              // data_size = 4 bytes
                | (1u << 20)                              // pad_enable
                | (5u << 22)                              // pad_interval: 64 DW
                | ((pad_amount_dw - 1u) << 25));          // pad_amount
    g1[1] = (int)(512u << 16);      // tensor_dim0 = 512 (bits 79:48 low half)
    g1[2] = (int)(512u << 16);      // dim0 hi=0 | tensor_dim1 = 512 (low half)
    g1[3] = (int)(tile_d0 << 16);   // dim1 hi=0 | tile_dim0
    g1[4] = (int)tile_d1;           // tile_dim1 | tile_dim2 = 0
    g1[5] = (int)stride_elts;       // tensor_dim0_stride[31:0]
    g1[6] = 0;                      // stride hi | tensor_dim1_stride lo
    g1[7] = 0;

    v4i z4 = {0, 0, 0, 0};
#if __clang_major__ >= 23
    v8i z8 = {0, 0, 0, 0, 0, 0, 0, 0};
    __builtin_amdgcn_tensor_load_to_lds(g0, g1, z4, z4, z8, 0);
#else
    __builtin_amdgcn_tensor_load_to_lds(g0, g1, z4, z4, 0);
#endif
}
#endif // USE_TDM

// ---------------------------------------------------------------------------
// Kernel 1: V_t[b][s] = sum_j q[j] * window(b, t+j)[s]
// window row i comes from x if i < 32, else from out[:, i-32, :].
// ---------------------------------------------------------------------------
__global__ __launch_bounds__(256)
void v_kernel(const float* __restrict__ x, const float* __restrict__ q,
              const float* __restrict__ out, float* __restrict__ V, int t) {
    __shared__ float qs[WINDOW];
    if (threadIdx.x < WINDOW) qs[threadIdx.x] = q[threadIdx.x];
    __syncthreads();

    int base = blockIdx.x * (256 * 4);
    #pragma unroll
    for (int k = 0; k < 4; ++k) {
        int idx = base + k * 256 + threadIdx.x;    // linear over BATCH*SIZE
        int b = idx >> 9;                           // / SIZE
        int s = idx & (SIZE - 1);
        float acc = 0.0f;
        #pragma unroll 1
        for (int j = 0; j < WINDOW; ++j) {
            int i = t + j;
            const float* src = (i < WINDOW)
                ? (x   + ((size_t)b * WINDOW  + i)            * SIZE + s)
                : (out + ((size_t)b * PLENGTH + (i - WINDOW)) * SIZE + s);
            acc = fmaf(qs[j], *src, acc);
        }
        V[idx] = acc;
    }
}

// ---------------------------------------------------------------------------
// Kernel 2: Y_t = V_t @ P  (M=256, N=512, K=512), Y written to out[:, t, :].
// 64x64 tile per workgroup; K in chunks of 64, double-buffered in LDS.
// Staging: TDM tensor loads (wave 0 issues, TENSORcnt) or async-b128 fallback.
// Compute: V_WMMA_F32_16X16X4_F32.
// ---------------------------------------------------------------------------
#define MT 64
#define NT 64
#define KC 64
#define NCHUNK (SIZE / KC)
#define AP (KC + 4)   // A pitch (68 fl): frag col reads hit 32 distinct banks
#define BP (NT + 8)   // B pitch (72 fl): half-wave rows in disjoint bank ranges

__device__ __forceinline__ void issue_chunk(const float* __restrict__ V,
                                            const float* __restrict__ P,
                                            int m0, int n0, int kc,
                                            float* asBuf, float* bsBuf, int tid) {
#if USE_TDM
    if (tid < 32) {                     // one wave drives the DMA engine
        tdm_load_2d(V + (size_t)m0 * SIZE + kc, asBuf, KC, MT, SIZE, AP - KC);
        tdm_load_2d(P + (size_t)kc * SIZE + n0, bsBuf, NT, KC, SIZE, BP - NT);
    }
#else
    #pragma unroll
    for (int i = tid; i < MT * (KC / 4); i += 256) {
        int r  = i >> 4;
        int c4 = (i & 15) << 2;
        async_cp16(V + (size_t)(m0 + r) * SIZE + kc + c4, &asBuf[r * AP + c4]);
    }
    #pragma unroll
    for (int i = tid; i < KC * (NT / 4); i += 256) {
        int r  = i >> 4;
        int c4 = (i & 15) << 2;
        async_cp16(P + (size_t)(kc + r) * SIZE + n0 + c4, &bsBuf[r * BP + c4]);
    }
#endif
}

__device__ __forceinline__ void wait_chunk(int outstanding, int tid) {
#if USE_TDM
    if (tid < 32) {
        if (outstanding)
            __builtin_amdgcn_s_wait_tensorcnt((short)2);
        else
            __builtin_amdgcn_s_wait_tensorcnt((short)0);
    }
#else
    if (outstanding)
        asm volatile("s_wait_asynccnt 0x8" ::: "memory");
    else
        asm volatile("s_wait_asynccnt 0x0" ::: "memory");
#endif
}

__global__ __launch_bounds__(256)
void gemm_kernel(const float* __restrict__ V, const float* __restrict__ P,
                 float* __restrict__ out, int t) {
    __shared__ alignas(16) float As[2][MT * AP];
    __shared__ alignas(16) float Bs[2][KC * BP];

    const int tid  = threadIdx.x;
    const int lane = tid & 31;
    const int wave = tid >> 5;
    const int mt   = wave & 3;     // M-tile within block (4 x 16 rows)
    const int ng   = wave >> 2;    // N-group within block (2 x 32 cols)
    const int m0   = blockIdx.y * MT;
    const int n0   = blockIdx.x * NT;

    const int lrow = lane & 15;
    const int koff = (lane >> 4) << 1;   // K-halfwave offset per ISA A/B layout

    v8f c0 = {};
    v8f c1 = {};

    // Prologue: start chunk 0 into buffer 0.
    issue_chunk(V, P, m0, n0, 0, As[0], Bs[0], tid);

    for (int c = 0; c < NCHUNK; ++c) {
        if (c + 1 < NCHUNK) {
            // Overlap: start chunk c+1 into the other buffer (its previous
            // consumers passed the end-of-iteration barrier of round c-1).
            issue_chunk(V, P, m0, n0, (c + 1) * KC, As[(c + 1) & 1],
                        Bs[(c + 1) & 1], tid);
            wait_chunk(1, tid);    // own chunk-c transfers have landed in LDS
        } else {
            wait_chunk(0, tid);
        }
        __syncthreads();           // all chunk-c data visible to all waves

        const float* as = As[c & 1];
        const float* bs = Bs[c & 1];
        #pragma unroll
        for (int k0 = 0; k0 < KC; k0 += 4) {
            // A fragment: 16x4 tile, lane half selects K {0,1} vs {2,3}
            const float* ap = &as[(mt * 16 + lrow) * AP + k0 + koff];
            v2f a; a.x = ap[0]; a.y = ap[1];
            // B fragments: 4x16 tiles for the two N sub-tiles
            const float* bp = &bs[(k0 + koff) * BP + ng * 32 + lrow];
            v2f b0; b0.x = bp[0];  b0.y = bp[BP];
            v2f b1; b1.x = bp[16]; b1.y = bp[16 + BP];

            c0 = __builtin_amdgcn_wmma_f32_16x16x4_f32(
                     false, a, false, b0, (short)0, c0, false, false);
            c1 = __builtin_amdgcn_wmma_f32_16x16x4_f32(
                     false, a, false, b1, (short)0, c1, false, false);
        }
        __syncthreads();   // protect buffer (c&1) before round c+1 overwrites it
    }

    // Store C: lane L holds col L%16; VGPR i is row i + 8*(L/16)  (ISA 7.12.2)
    const int row0 = m0 + mt * 16 + (lane >> 4) * 8;   // batch index base
    const int col  = n0 + ng * 32 + lrow;
    float* o = out + ((size_t)row0 * PLENGTH + t) * SIZE + col;
    const size_t bstride = (size_t)PLENGTH * SIZE;     // next batch row
    #pragma unroll
    for (int i = 0; i < 8; ++i) {
        o[(size_t)i * bstride]      = c0[i];
        o[(size_t)i * bstride + 16] = c1[i];
    }
}

// ---------------------------------------------------------------------------
extern "C" void kernel_launch(void* const* d_in, const int* in_sizes, int n_in,
                              void* d_out, int out_size, void* d_ws, size_t ws_size,
                              hipStream_t stream) {
    const float* x = (const float*)d_in[0];   // (256, 32, 512)
    const float* q = (const float*)d_in[1];   // (1, 32)
    const float* P = (const float*)d_in[2];   // (512, 512)
    float* out = (float*)d_out;               // (256, 128, 512)
    float* V   = (float*)d_ws;                // scratch: 256*512 floats = 512 KB

    dim3 gemm_grid(SIZE / NT, BATCH / MT);    // (8, 4)

    for (int t = 0; t < PLENGTH; ++t) {
        v_kernel<<<(BATCH * SIZE) / (256 * 4), 256, 0, stream>>>(x, q, out, V, t);
        gemm_kernel<<<gemm_grid, 256, 0, stream>>>(V, P, out, t);
    }
}